// NEP_75058848465313
// MI455X (gfx1250) — compile-verified
//
#include <hip/hip_runtime.h>
#include <math.h>

#define MNEI   200
#define BATCH  4
#define NATOM  2048
#define NROW   (BATCH*NATOM)   // 8192
#define NFEAT  35
#define NFP    36
#define HID    100
#define KCH    9
#define PI_F   3.14159265358979323846f

__constant__ float C3B_[24]  = {0.238732414637843f,0.119366207318922f,0.119366207318922f,0.099471839432435f,
                                0.596831036594608f,0.596831036594608f,0.149207759148652f,0.149207759148652f,
                                0.139260575205408f,0.104445431404056f,0.104445431404056f,1.044454314040563f,
                                1.044454314040563f,0.174075719006761f,0.174075719006761f,0.011190581936149f,
                                0.223811638722978f,0.223811638722978f,0.111905819361489f,0.111905819361489f,
                                1.566681471060845f,1.566681471060845f,0.195835183882606f,0.195835183882606f};
__constant__ float C3BM_[24] = {1,2,2,1,2,2,2,2,1,2,2,2,2,2,2,1,2,2,2,2,2,2,2,2};
__constant__ float C4B_[5]   = {-0.007499480826664f,-0.134990654879954f,0.067495327439977f,
                                 0.404971964639861f,-0.809943929279723f};
__constant__ float C5B_[3]   = {0.026596810706114f,0.053193621412227f,0.026596810706114f};

__device__ __forceinline__ unsigned f2key(float f) {
  unsigned b = __float_as_uint(f);
  return (b & 0x80000000u) ? ~b : (b | 0x80000000u);
}
__device__ __forceinline__ float key2f(unsigned k) {
  unsigned b = (k & 0x80000000u) ? (k & 0x7fffffffu) : ~k;
  return __uint_as_float(b);
}
__device__ __forceinline__ float wred(float v) {
  #pragma unroll
  for (int o = 16; o > 0; o >>= 1) v += __shfl_xor(v, o, 32);
  return v;
}

// -------------------- kernel 0: init min/max keys --------------------
__global__ void nep_init_kernel(unsigned* __restrict__ minmax) {
  int t = threadIdx.x;
  if (t < NFEAT)            minmax[t] = 0xFFFFFFFFu;   // min slots
  else if (t < 2 * NFEAT)   minmax[t] = 0u;            // max slots
}

// -------------------- kernel 1: per-atom features --------------------
// 1 wave per atom, 8 atoms (waves) per 256-thread block.
__global__ __launch_bounds__(256) void nep_feats_kernel(
    const int*   __restrict__ list_neigh,
    const int*   __restrict__ typemap,
    const float* __restrict__ ImageDR,     // (B,N,M,4)
    const float* __restrict__ c2,          // (2,2,5,9)
    const float* __restrict__ c3,          // (2,2,5,9)
    float*       __restrict__ feats,       // (NROW, NFP)
    unsigned*    __restrict__ minmax)      // [35 min | 35 max]
{
  __shared__ float lds_c3[8 * 90];
  __shared__ float lds_f[8 * NFP];

  const int lane = threadIdx.x & 31;
  const int wv   = threadIdx.x >> 5;
  const int atom = blockIdx.x * 8 + wv;          // = b*NATOM + i
  const int ia   = atom & (NATOM - 1);
  const int ti   = typemap[ia];

  for (int e = lane; e < 90; e += 32)            // stage c3[ti][*][*][*]
    lds_c3[wv * 90 + e] = c3[ti * 90 + e];
  __syncthreads();

  float F0[KCH], F1[KCH], s[5][24];
  #pragma unroll
  for (int k = 0; k < KCH; k++) { F0[k] = 0.f; F1[k] = 0.f; }
  #pragma unroll
  for (int n = 0; n < 5; n++)
    #pragma unroll
    for (int c = 0; c < 24; c++) s[n][c] = 0.f;

  const float* ccw = &lds_c3[wv * 90];

  for (int m = lane; m < MNEI; m += 32) {
    const long idx = (long)atom * MNEI + m;
    int   lv = list_neigh[idx];
    float4 dr = ((const float4*)ImageDR)[idx];
    float d  = dr.x;
    float validf = (lv > 0) ? 1.f : 0.f;
    float dinv = (d > 0.f) ? 1.f / fmaxf(d, 1e-12f) : 0.f;
    float x = dr.y * dinv, y = dr.z * dinv, z = dr.w * dinv;
    int tj = (m >= 100) ? 1 : 0;

    // ---- radial Chebyshev (rc=6) ----
    {
      float fc = 0.5f * (cosf(PI_F * d / 6.f) + 1.f) * ((d < 6.f) ? 1.f : 0.f) * validf;
      float xr = d / 6.f - 1.f; xr = 2.f * xr * xr - 1.f;
      float fk[KCH];
      float t0 = 1.f, t1 = xr;
      fk[0] = 0.5f * (t0 + 1.f) * fc;
      fk[1] = 0.5f * (t1 + 1.f) * fc;
      #pragma unroll
      for (int k = 2; k < KCH; k++) {
        float t2 = 2.f * xr * t1 - t0;
        fk[k] = 0.5f * (t2 + 1.f) * fc;
        t0 = t1; t1 = t2;
      }
      float m0 = (tj == 0) ? 1.f : 0.f, m1 = 1.f - m0;
      #pragma unroll
      for (int k = 0; k < KCH; k++) { F0[k] += fk[k] * m0; F1[k] += fk[k] * m1; }
    }

    // ---- angular Chebyshev (rc=4) ----
    float fka[KCH];
    {
      float fc = 0.5f * (cosf(PI_F * d / 4.f) + 1.f) * ((d < 4.f) ? 1.f : 0.f) * validf;
      float xa = d / 4.f - 1.f; xa = 2.f * xa * xa - 1.f;
      float t0 = 1.f, t1 = xa;
      fka[0] = 0.5f * (t0 + 1.f) * fc;
      fka[1] = 0.5f * (t1 + 1.f) * fc;
      #pragma unroll
      for (int k = 2; k < KCH; k++) {
        float t2 = 2.f * xa * t1 - t0;
        fka[k] = 0.5f * (t2 + 1.f) * fc;
        t0 = t1; t1 = t2;
      }
    }

    // ---- blm ----
    float x2 = x * x, y2 = y * y, z2 = z * z, xmy = x2 - y2;
    float blm[24];
    blm[0]=z; blm[1]=x; blm[2]=y;
    blm[3]=3.f*z2-1.f; blm[4]=x*z; blm[5]=y*z; blm[6]=xmy; blm[7]=2.f*x*y;
    blm[8]=(5.f*z2-3.f)*z; blm[9]=(5.f*z2-1.f)*x; blm[10]=(5.f*z2-1.f)*y;
    blm[11]=xmy*z; blm[12]=2.f*x*y*z; blm[13]=(x2-3.f*y2)*x; blm[14]=(3.f*x2-y2)*y;
    blm[15]=(35.f*z2-30.f)*z2+3.f; blm[16]=(7.f*z2-3.f)*x*z; blm[17]=(7.f*z2-3.f)*y*z;
    blm[18]=(7.f*z2-1.f)*xmy; blm[19]=(7.f*z2-1.f)*2.f*x*y;
    blm[20]=(x2-3.f*y2)*x*z; blm[21]=(3.f*x2-y2)*y*z;
    blm[22]=xmy*xmy-4.f*x2*y2; blm[23]=4.f*x*y*xmy;

    const float* cc = ccw + tj * 45;
    #pragma unroll
    for (int n = 0; n < 5; n++) {
      float g = 0.f;
      #pragma unroll
      for (int k = 0; k < KCH; k++) g += fka[k] * cc[n * KCH + k];
      #pragma unroll
      for (int c = 0; c < 24; c++) s[n][c] += g * blm[c];
    }
  }

  // ---- cross-lane reduction ----
  #pragma unroll
  for (int k = 0; k < KCH; k++) { F0[k] = wred(F0[k]); F1[k] = wred(F1[k]); }
  #pragma unroll
  for (int n = 0; n < 5; n++)
    #pragma unroll
    for (int c = 0; c < 24; c++) s[n][c] = wred(s[n][c]);

  if (lane == 0) {
    float feat[NFP];
    const float* c2b = &c2[ti * 90];
    #pragma unroll
    for (int n = 0; n < 5; n++) {
      float q = 0.f;
      #pragma unroll
      for (int k = 0; k < KCH; k++)
        q += c2b[n * KCH + k] * F0[k] + c2b[45 + n * KCH + k] * F1[k];
      feat[n] = q;
    }
    #pragma unroll
    for (int n = 0; n < 5; n++) {
      float g0 = 0.f, g1 = 0.f, g2 = 0.f, g3 = 0.f;
      #pragma unroll
      for (int c = 0; c < 3; c++)   g0 += s[n][c] * s[n][c] * C3B_[c] * C3BM_[c];
      #pragma unroll
      for (int c = 3; c < 8; c++)   g1 += s[n][c] * s[n][c] * C3B_[c] * C3BM_[c];
      #pragma unroll
      for (int c = 8; c < 15; c++)  g2 += s[n][c] * s[n][c] * C3B_[c] * C3BM_[c];
      #pragma unroll
      for (int c = 15; c < 24; c++) g3 += s[n][c] * s[n][c] * C3B_[c] * C3BM_[c];
      feat[5 + n]  = g0; feat[10 + n] = g1; feat[15 + n] = g2; feat[20 + n] = g3;

      float s3 = s[n][3], s4 = s[n][4], s5 = s[n][5], s6 = s[n][6], s7 = s[n][7];
      feat[25 + n] = C4B_[0]*s3*s3*s3 + C4B_[1]*s3*(s4*s4+s5*s5)
                   + C4B_[2]*s3*(s6*s6+s7*s7) + C4B_[3]*s6*(s5*s5-s4*s4)
                   + C4B_[4]*s4*s5*s7;

      float s0 = s[n][0], s1 = s[n][1], s2 = s[n][2];
      float s0sq = s0*s0, sp = s1*s1 + s2*s2;
      feat[30 + n] = C5B_[0]*s0sq*s0sq + C5B_[1]*s0sq*sp + C5B_[2]*sp*sp;
    }
    feat[NFEAT] = 0.f;
    #pragma unroll
    for (int j = 0; j < NFP; j++) lds_f[wv * NFP + j] = feat[j];
  }
  __syncthreads();

  // coalesced store of the 8 feature rows
  for (int e = threadIdx.x; e < 8 * NFP; e += 256)
    feats[(long)blockIdx.x * 8 * NFP + e] = lds_f[e];

  // block-level min/max then global atomics (ordered-uint keys)
  if (threadIdx.x < NFEAT) {
    int f = threadIdx.x;
    float mn = lds_f[f], mx = lds_f[f];
    #pragma unroll
    for (int a = 1; a < 8; a++) {
      float v = lds_f[a * NFP + f];
      mn = fminf(mn, v); mx = fmaxf(mx, v);
    }
    atomicMin(&minmax[f],         f2key(mn));
    atomicMax(&minmax[NFEAT + f], f2key(mx));
  }
}

// -------------------- kernel 2: scaler --------------------
__global__ void nep_scaler_kernel(const unsigned* __restrict__ minmax,
                                  float* __restrict__ scaler) {
  int f = threadIdx.x;
  if (f < NFP) {
    float sc = 0.f;
    if (f < NFEAT) {
      float mn = key2f(minmax[f]);
      float mx = key2f(minmax[NFEAT + f]);
      sc = 1.f / (mx - mn);
    }
    scaler[f] = sc;
  }
}

// -------------------- kernel 3: MLP via fp32 WMMA --------------------
typedef __attribute__((ext_vector_type(2))) float v2f;
typedef __attribute__((ext_vector_type(8))) float v8f;

__global__ __launch_bounds__(32) void nep_mlp_kernel(
    const float* __restrict__ feats,
    const float* __restrict__ scaler,
    const int*   __restrict__ typemap,
    const float* __restrict__ W0,   // (2,35,100)
    const float* __restrict__ b0,   // (2,100)
    const float* __restrict__ W1,   // (2,100,1)
    const float* __restrict__ b1,   // (2,1)
    float*       __restrict__ out)  // [Etot(4) | Ei(8192)]
{
  const int lane = threadIdx.x;
  const int tile = blockIdx.x;           // 512 tiles of 16 rows
  const int r16  = lane & 15;
  const int hi   = lane >> 4;

  int   ty[8];
  float rsum[8];
  #pragma unroll
  for (int g = 0; g < 8; g++) {
    int grow = tile * 16 + g + 8 * hi;
    ty[g]   = typemap[grow & (NATOM - 1)];
    rsum[g] = 0.f;
  }

  for (int ct = 0; ct < 7; ct++) {       // 7 col tiles -> 112 >= HID
    v8f acc0 = {}; v8f acc1 = {};
    int   col   = ct * 16 + r16;
    int   colc  = (col < HID) ? col : (HID - 1);
    float cmask = (col < HID) ? 1.f : 0.f;

    #pragma unroll
    for (int kt = 0; kt < KCH; kt++) {   // K = 36 in steps of 4
      int k = kt * 4 + hi * 2;           // this lane's K pair: k, k+1 (<= 35)
      int arow = tile * 16 + r16;
      v2f a;
      a.x = feats[arow * NFP + k]     * scaler[k];
      a.y = feats[arow * NFP + k + 1] * scaler[k + 1];

      int   k0c = (k     < NFEAT) ? k     : (NFEAT - 1);
      int   k1c = (k + 1 < NFEAT) ? k + 1 : (NFEAT - 1);
      float km0 = (k     < NFEAT) ? 1.f : 0.f;
      float km1 = (k + 1 < NFEAT) ? 1.f : 0.f;

      v2f bw0, bw1;
      bw0.x = W0[          k0c * HID + colc] * km0 * cmask;
      bw0.y = W0[          k1c * HID + colc] * km1 * cmask;
      bw1.x = W0[NFEAT*HID + k0c * HID + colc] * km0 * cmask;
      bw1.y = W0[NFEAT*HID + k1c * HID + colc] * km1 * cmask;

      acc0 = __builtin_amdgcn_wmma_f32_16x16x4_f32(false, a, false, bw0,
                                                   (short)0, acc0, false, false);
      acc1 = __builtin_amdgcn_wmma_f32_16x16x4_f32(false, a, false, bw1,
                                                   (short)0, acc1, false, false);
    }

    #pragma unroll
    for (int g = 0; g < 8; g++) {
      int   t  = ty[g];
      float v  = t ? acc1[g] : acc0[g];
      float hb = v + b0[t * HID + colc] * cmask;
      float hv = tanhf(hb);
      float p  = hv * W1[t * HID + colc] * cmask;
      #pragma unroll
      for (int off = 8; off > 0; off >>= 1) p += __shfl_xor(p, off, 32);
      rsum[g] += p;
    }
  }

  if (r16 == 0) {
    #pragma unroll
    for (int g = 0; g < 8; g++) {
      int grow = tile * 16 + g + 8 * hi;
      out[4 + grow] = rsum[g] + b1[ty[g]];
    }
  }
}

// -------------------- kernel 4: Etot reduction --------------------
__global__ __launch_bounds__(256) void nep_etot_kernel(float* __restrict__ out) {
  __shared__ float red[256];
  int b = blockIdx.x, t = threadIdx.x;
  float sum = 0.f;
  for (int j = t; j < NATOM; j += 256) sum += out[4 + b * NATOM + j];
  red[t] = sum;
  __syncthreads();
  for (int w = 128; w > 0; w >>= 1) {
    if (t < w) red[t] += red[t + w];
    __syncthreads();
  }
  if (t == 0) out[b] = red[0];
}

// -------------------- host launch --------------------
extern "C" void kernel_launch(void* const* d_in, const int* in_sizes, int n_in,
                              void* d_out, int out_size, void* d_ws, size_t ws_size,
                              hipStream_t stream) {
  const int*   list_neigh = (const int*)d_in[0];
  const int*   typemap    = (const int*)d_in[1];
  const float* ImageDR    = (const float*)d_in[3];
  const float* c2         = (const float*)d_in[5];
  const float* c3         = (const float*)d_in[6];
  const float* W0         = (const float*)d_in[7];
  const float* b0         = (const float*)d_in[8];
  const float* W1         = (const float*)d_in[9];
  const float* b1         = (const float*)d_in[10];
  float* out = (float*)d_out;

  float*    wsf    = (float*)d_ws;
  float*    feats  = wsf;                              // NROW*NFP floats
  unsigned* minmax = (unsigned*)(wsf + NROW * NFP);    // 70 u32
  float*    scaler = wsf + NROW * NFP + 128;           // 36 floats

  nep_init_kernel<<<1, 96, 0, stream>>>(minmax);
  nep_feats_kernel<<<NROW / 8, 256, 0, stream>>>(list_neigh, typemap, ImageDR,
                                                 c2, c3, feats, minmax);
  nep_scaler_kernel<<<1, 64, 0, stream>>>(minmax, scaler);
  nep_mlp_kernel<<<NROW / 16, 32, 0, stream>>>(feats, scaler, typemap,
                                               W0, b0, W1, b1, out);
  nep_etot_kernel<<<BATCH, 256, 0, stream>>>(out);
}